// DropBlock_86517821213022
// MI455X (gfx1250) — compile-verified
//
#include <hip/hip_runtime.h>
#include <hip/hip_bf16.h>

// ---------------- problem constants (from reference) ----------------
#define BB 64
#define CC 256
#define HH 56
#define WW 56
#define BSZ 5
#define HM (HH - (BSZ - 1))          // 52
#define WM (WW - (BSZ - 1))          // 52
#define NPLANES (BB * CC)            // 16384
#define MELEMS (HM * WM)             // 2704 int32 per mask plane (10816 B)
#define MUNITS (MELEMS * 4 / 16)     // 676 x 16B units (exact)
#define OELEMS (HH * WW)             // 3136 per output plane
#define PPB 8                        // planes per block
#define NBLK (NPLANES / PPB)         // 2048
#define NTHREADS 256
#define COUNT_M ((unsigned)NPLANES * (unsigned)OELEMS)  // 51,380,224

// ---------------- CDNA5 async global->LDS path ----------------
#ifndef __has_builtin
#define __has_builtin(x) 0
#endif

#if defined(__HIP_DEVICE_COMPILE__) && \
    __has_builtin(__builtin_amdgcn_global_load_async_to_lds_b128)
#define ASYNC_LDS 1
#endif

#if defined(__HIP_DEVICE_COMPILE__) && \
    __has_builtin(__builtin_amdgcn_s_wait_asynccnt)
#define WAIT_ASYNC(n) __builtin_amdgcn_s_wait_asynccnt(n)
#elif defined(ASYNC_LDS)
#define WAIT_ASYNC(n) asm volatile("s_wait_asynccnt %0" ::"i"(n) : "memory")
#else
#define WAIT_ASYNC(n) ((void)0)
#endif

// Exact vector type the builtin expects: v4i (vector_size form), AS1 src / AS3 dst.
typedef int vi4 __attribute__((vector_size(16)));

__device__ __forceinline__ void async_ld16(const int4* g, int4* l) {
#ifdef ASYNC_LDS
  __builtin_amdgcn_global_load_async_to_lds_b128(
      (__attribute__((address_space(1))) vi4*)g,
      (__attribute__((address_space(3))) vi4*)l,
      /*imm offset*/ 0, /*cpol*/ 0);
#else
  *l = *g;  // synchronous fallback
#endif
}

// Copy one 52x52 int32 mask plane (10816B = 676 x 16B) into LDS.
// Every lane issues exactly 3 b128 ops (index clamped -> full EXEC,
// uniform per-wave ASYNCcnt of 3 per copy).
__device__ __forceinline__ void copy_plane_async(const int* __restrict__ gsrc,
                                                 int* __restrict__ ldst) {
  const int4* g = (const int4*)gsrc;
  int4* l = (int4*)ldst;
#pragma unroll
  for (int k = 0; k < 3; ++k) {
    int u = (int)threadIdx.x + k * NTHREADS;
    if (u > MUNITS - 1) u = MUNITS - 1;  // duplicate tail, harmless
    async_ld16(g + u, l + u);
  }
}

// ---------------- pass A: dilation + global count of ones ----------------
__global__ __launch_bounds__(NTHREADS) void dropblock_sum(
    const int* __restrict__ mask, unsigned* __restrict__ counter) {
  __shared__ int mbuf[2][MELEMS];
  __shared__ int rbuf[HM * WW];  // row-dilated: 52 rows x 56 cols
  __shared__ unsigned s_dil;

  if (threadIdx.x == 0) s_dil = 0u;

  const int plane0 = blockIdx.x * PPB;
  copy_plane_async(mask + (size_t)plane0 * MELEMS, mbuf[0]);

  for (int p = 0; p < PPB; ++p) {
    if (p + 1 < PPB) {
      copy_plane_async(mask + (size_t)(plane0 + p + 1) * MELEMS,
                       mbuf[(p + 1) & 1]);
      WAIT_ASYNC(3);  // only the freshly-issued prefetch may remain
    } else {
      WAIT_ASYNC(0);
    }
    __syncthreads();

    const int* m = mbuf[p & 1];
    // Row stage: rbuf[r][w] = OR over mask[r][w-4..w] clipped to [0,51]
    for (int idx = threadIdx.x; idx < HM * WW; idx += NTHREADS) {
      int r = idx / WW, w = idx - r * WW;
      int lo = w - (BSZ - 1); if (lo < 0) lo = 0;
      int hi = w;             if (hi > WM - 1) hi = WM - 1;
      int v = 0;
      for (int j = lo; j <= hi; ++j) v |= m[r * WM + j];
      rbuf[idx] = v;
    }
    __syncthreads();

    // Column stage: dilated[h][w] = OR over rbuf[h-4..h][w]; count ones
    unsigned cnt = 0;
    for (int idx = threadIdx.x; idx < OELEMS; idx += NTHREADS) {
      int h = idx / WW, w = idx - h * WW;
      int lo = h - (BSZ - 1); if (lo < 0) lo = 0;
      int hi = h;             if (hi > HM - 1) hi = HM - 1;
      int v = 0;
      for (int i = lo; i <= hi; ++i) v |= rbuf[i * WW + w];
      cnt += (unsigned)v;
    }
    atomicAdd(&s_dil, cnt);
    __syncthreads();  // rbuf + mbuf[p&1] free for reuse; s_dil settled
  }

  if (threadIdx.x == 0)
    atomicAdd(counter, (unsigned)(PPB * OELEMS) - s_dil);  // ones = total - dilated
}

// ---------------- pass B: recompute dilation, fused scaled multiply ----------------
__global__ __launch_bounds__(NTHREADS) void dropblock_apply(
    const float* __restrict__ x, const int* __restrict__ mask,
    const unsigned* __restrict__ counter, float* __restrict__ out) {
  __shared__ int mbuf[2][MELEMS];
  __shared__ int rbuf[HM * WW];

  const float scale = (float)COUNT_M / (float)(*counter);

  const int plane0 = blockIdx.x * PPB;
  copy_plane_async(mask + (size_t)plane0 * MELEMS, mbuf[0]);

  for (int p = 0; p < PPB; ++p) {
    if (p + 1 < PPB) {
      copy_plane_async(mask + (size_t)(plane0 + p + 1) * MELEMS,
                       mbuf[(p + 1) & 1]);
      WAIT_ASYNC(3);
    } else {
      WAIT_ASYNC(0);
    }
    __syncthreads();

    const int* m = mbuf[p & 1];
    for (int idx = threadIdx.x; idx < HM * WW; idx += NTHREADS) {
      int r = idx / WW, w = idx - r * WW;
      int lo = w - (BSZ - 1); if (lo < 0) lo = 0;
      int hi = w;             if (hi > WM - 1) hi = WM - 1;
      int v = 0;
      for (int j = lo; j <= hi; ++j) v |= m[r * WM + j];
      rbuf[idx] = v;
    }
    __syncthreads();

    const size_t base = (size_t)(plane0 + p) * OELEMS;
    for (int idx = threadIdx.x; idx < OELEMS; idx += NTHREADS) {
      int h = idx / WW, w = idx - h * WW;
      int lo = h - (BSZ - 1); if (lo < 0) lo = 0;
      int hi = h;             if (hi > HM - 1) hi = HM - 1;
      int v = 0;
      for (int i = lo; i <= hi; ++i) v |= rbuf[i * WW + w];
      // x read-once / out write-once -> non-temporal, keep L2 for mask
      float xv = __builtin_nontemporal_load(x + base + idx);
      float r = v ? 0.0f : (scale * xv);
      __builtin_nontemporal_store(r, out + base + idx);
    }
    __syncthreads();
  }
}

__global__ void dropblock_zero(unsigned* c) {
  if (threadIdx.x == 0) *c = 0u;
}

extern "C" void kernel_launch(void* const* d_in, const int* in_sizes, int n_in,
                              void* d_out, int out_size, void* d_ws, size_t ws_size,
                              hipStream_t stream) {
  const float* x   = (const float*)d_in[0];
  const int* mask  = (const int*)d_in[1];
  // d_in[2] = block_size scalar (compile-time 5 here)
  unsigned* counter = (unsigned*)d_ws;
  float* out = (float*)d_out;

  dropblock_zero<<<1, 64, 0, stream>>>(counter);
  dropblock_sum<<<NBLK, NTHREADS, 0, stream>>>(mask, counter);
  dropblock_apply<<<NBLK, NTHREADS, 0, stream>>>(x, mask, counter, out);
}